// Mamba3Clean_6236292514072
// MI455X (gfx1250) — compile-verified
//
#include <hip/hip_runtime.h>
#include <hip/hip_bf16.h>
#include <math.h>

// ---------------- problem constants ----------------
#define SEQLEN   1024
#define D_MODEL  1024
#define D_INNER  2048
#define D_STATE  128
#define NHEADS   32
#define HEADDIM  64
#define RROT     32          // R = rotary half-split
#define DPROJ    4480        // 2*2048 + 2*128 + 3*32 + 32
// proj row layout offsets
#define OFF_Z    0
#define OFF_X    2048
#define OFF_B    4096
#define OFF_C    4224
#define OFF_DDT  4352
#define OFF_DDA  4384
#define OFF_TRAP 4416
#define OFF_ANG  4448
// LDS stage buffer layout (floats): [0:128)=B, [128:256)=C, [256:320)=x, [320:352)=rc, [352:384)=rs
#define ST_B   0
#define ST_C   128
#define ST_X   256
#define ST_RC  320
#define ST_RS  352
#define ST_LEN 384

typedef float v2f __attribute__((ext_vector_type(2)));
typedef float v8f __attribute__((ext_vector_type(8)));

#if __has_builtin(__builtin_amdgcn_global_load_async_to_lds_b32)
#define USE_ASYNC_LDS 1
typedef __attribute__((address_space(1))) int as1_int;   // prints as "__device__ int*"
typedef __attribute__((address_space(3))) int as3_int;   // prints as "__shared__ int*"
__device__ __forceinline__ void async_ld32(float* lds_dst, const float* gsrc) {
  __builtin_amdgcn_global_load_async_to_lds_b32(
      (as1_int*)gsrc, (as3_int*)lds_dst, 0, 0);
}
#endif

#if __has_builtin(__builtin_amdgcn_s_wait_asynccnt)
#define WAIT_ASYNC() __builtin_amdgcn_s_wait_asynccnt(0)
#else
#define WAIT_ASYNC() asm volatile("s_wait_asynccnt 0x0" ::: "memory")
#endif

__device__ __forceinline__ float softplus_f(float x) {
  return (x > 20.f) ? x : log1pf(expf(x));
}

// ======================================================================
// Kernel 1/5: fp32 WMMA GEMM, C[M,N] = A[M,K] @ B[N,K]^T  (TN layout)
// One wave computes one 16x16 C tile via V_WMMA_F32_16X16X4_F32.
// A-frag layout (32b A 16x4): lanes 0-15 hold M=lane,K={k,k+1};
//                             lanes 16-31 hold M=lane-16,K={k+2,k+3}.
// B-frag mirrors with N=lane&15. C/D: VGPR i -> M = i + 8*(lane>>4), N = lane&15.
// ======================================================================
__global__ void wmma_gemm_tn(const float* __restrict__ A,
                             const float* __restrict__ B,
                             float* __restrict__ C,
                             int K, int N) {
  const int lane = threadIdx.x & 31;
  const int wave = threadIdx.x >> 5;
  const int half = lane >> 4;
  const int l    = lane & 15;
  const int m0 = blockIdx.x * 16;
  const int n0 = (blockIdx.y * 4 + wave) * 16;

  const float* arow = A + (size_t)(m0 + l) * (size_t)K;
  const float* brow = B + (size_t)(n0 + l) * (size_t)K;

  v8f acc = {};
#pragma unroll 4
  for (int k = 0; k < K; k += 4) {
    const int kk = k + 2 * half;
    v2f a = *(const v2f*)(arow + kk);   // global_load_b64
    v2f b = *(const v2f*)(brow + kk);   // global_load_b64
    acc = __builtin_amdgcn_wmma_f32_16x16x4_f32(
        /*neg_a=*/false, a, /*neg_b=*/false, b,
        /*c_mod=*/(short)0, acc, /*reuse_a=*/false, /*reuse_b=*/false);
  }
#pragma unroll
  for (int i = 0; i < 8; ++i) {
    const int row = m0 + i + 8 * half;
    C[(size_t)row * (size_t)N + (size_t)(n0 + l)] = acc[i];
  }
}

// ======================================================================
// Kernel 2/5: per-(t,h) gate scalars: dt, alpha, beta, gamma
// ======================================================================
__global__ void prep_scalars(const float* __restrict__ proj,
                             const float* __restrict__ dt_bias,
                             float* __restrict__ dtb,
                             float* __restrict__ alphab,
                             float* __restrict__ betab,
                             float* __restrict__ gammab) {
  const int i = blockIdx.x * blockDim.x + threadIdx.x;   // i = t*32 + h
  if (i >= SEQLEN * NHEADS) return;
  const int t = i >> 5, h = i & 31;
  const float* row = proj + (size_t)t * DPROJ;
  const float dd_dt = row[OFF_DDT + h];
  const float dd_A  = row[OFF_DDA + h];
  const float trap  = row[OFF_TRAP + h];
  const float dt = softplus_f(dd_dt + dt_bias[h]);
  const float Ah = fminf(-softplus_f(dd_A), -1.0e-4f);
  const float tr = 1.f / (1.f + expf(-trap));
  const float alpha = expf(Ah * dt);
  dtb[i]    = dt;
  alphab[i] = alpha;
  betab[i]  = (1.f - tr) * dt * alpha;
  gammab[i] = tr * dt;
}

// ======================================================================
// Kernel 3/5: theta cumsum over time per (h,r); store cos/sin directly.
// 1024 independent serial chains, one thread each.
// ======================================================================
__global__ void theta_kernel(const float* __restrict__ proj,
                             const float* __restrict__ dtb,
                             float* __restrict__ cosb,
                             float* __restrict__ sinb) {
  const int i = blockIdx.x * blockDim.x + threadIdx.x;   // i = h*32 + r
  if (i >= NHEADS * RROT) return;
  const int h = i >> 5, r = i & 31;
  float th = 0.f;
  for (int t = 0; t < SEQLEN; ++t) {
    const float ang = proj[(size_t)t * DPROJ + OFF_ANG + r];
    const float dt  = dtb[t * NHEADS + h];
    th += tanhf(ang) * dt * 3.14159265358979323846f;
    float c, s;
    sincosf(th, &s, &c);
    cosb[t * (NHEADS * RROT) + i] = c;
    sinb[t * (NHEADS * RROT) + i] = s;
  }
}

// ======================================================================
// Kernel 4/5: the sequential scan. One workgroup per head.
// 256 threads = 64 d-lanes x 4 n-groups; thread (d,g) owns n = g + 4j,
// j = 0..31 (rotary part is j<8: compile-time, no divergence).
// State sr[32]/si[8] kept in VGPRs. Per-step operands (B/C/x/cos/sin) are
// DOUBLE-BUFFERED in LDS and prefetched with GLOBAL_LOAD_ASYNC_TO_LDS_B32
// (ASYNCcnt + one barrier per step) so the global-load latency of step t+1
// overlaps the ~200 VALU ops of step t -- the scan is the only sequential
// part of the model and is pure latency, not bandwidth.
// Fuses y + D*x and SiLU(z) gating into the epilogue -> writes act[t, 2048].
// ======================================================================
__device__ __forceinline__ void stage_step(float* __restrict__ buf,
                                           const float* __restrict__ row,
                                           const float* __restrict__ cost,
                                           const float* __restrict__ sint,
                                           int tid, int h) {
#if defined(USE_ASYNC_LDS)
  if (tid < 128)      async_ld32(buf + ST_B + tid,        row + OFF_B + tid);
  else                async_ld32(buf + ST_C + (tid-128),  row + OFF_C + (tid - 128));
  if (tid < 64)       async_ld32(buf + ST_X + tid,        row + OFF_X + h * HEADDIM + tid);
  else if (tid < 96)  async_ld32(buf + ST_RC + (tid-64),  cost + h * RROT + (tid - 64));
  else if (tid < 128) async_ld32(buf + ST_RS + (tid-96),  sint + h * RROT + (tid - 96));
#else
  if (tid < 128)      buf[ST_B + tid]        = row[OFF_B + tid];
  else                buf[ST_C + (tid-128)]  = row[OFF_C + (tid - 128)];
  if (tid < 64)       buf[ST_X + tid]        = row[OFF_X + h * HEADDIM + tid];
  else if (tid < 96)  buf[ST_RC + (tid-64)]  = cost[h * RROT + (tid - 64)];
  else if (tid < 128) buf[ST_RS + (tid-96)]  = sint[h * RROT + (tid - 96)];
#endif
}

__global__ void __launch_bounds__(256)
scan_kernel(const float* __restrict__ proj,
            const float* __restrict__ alphab,
            const float* __restrict__ betab,
            const float* __restrict__ gammab,
            const float* __restrict__ cosb,
            const float* __restrict__ sinb,
            const float* __restrict__ Dvec,
            float* __restrict__ act) {
  const int h   = blockIdx.x;
  const int tid = threadIdx.x;
  const int d   = tid >> 2;
  const int g   = tid & 3;

  __shared__ float stage[2][ST_LEN];

  float sr[32];
  float si[8];
#pragma unroll
  for (int j = 0; j < 32; ++j) sr[j] = 0.f;
#pragma unroll
  for (int j = 0; j < 8; ++j)  si[j] = 0.f;
  float px = 0.f;
  const float Dh = Dvec[h];

  // prologue: stage step 0
  stage_step(stage[0], proj, cosb, sinb, tid, h);
  WAIT_ASYNC();
  __syncthreads();

  for (int t = 0; t < SEQLEN; ++t) {
    // prefetch step t+1 into the other buffer (in flight during compute)
    if (t + 1 < SEQLEN) {
      stage_step(stage[(t + 1) & 1],
                 proj + (size_t)(t + 1) * DPROJ,
                 cosb + (size_t)(t + 1) * (NHEADS * RROT),
                 sinb + (size_t)(t + 1) * (NHEADS * RROT),
                 tid, h);
    }
    const float* __restrict__ S = stage[t & 1];

    const float alpha = alphab[t * NHEADS + h];   // uniform -> scalar loads
    const float beta  = betab[t * NHEADS + h];
    const float gamma = gammab[t * NHEADS + h];
    const float x = S[ST_X + d];
    const float v = px * beta + x * gamma;

    float y = 0.f;
#pragma unroll
    for (int j = 0; j < 32; ++j) {
      const int n = g + 4 * j;
      const float base = S[ST_B + n] * v;
      if (j < 8) {                     // n < R: rotary slice (compile-time)
        const float rc = S[ST_RC + n], rs = S[ST_RS + n];
        sr[j] = sr[j] * alpha + base * rc;
        si[j] = si[j] * alpha + base * rs;
        y += (sr[j] * rc + si[j] * rs) * S[ST_C + n];
      } else {
        sr[j] = sr[j] * alpha + base;
        y += sr[j] * S[ST_C + n];
      }
    }
    // reduce across the 4 n-groups (consecutive wave32 lanes)
    y += __shfl_xor(y, 1, 32);
    y += __shfl_xor(y, 2, 32);
    if (g == 0) {
      const float yv = y + Dh * x;
      const float z  = proj[(size_t)t * DPROJ + OFF_Z + h * HEADDIM + d];
      const float sz = z / (1.f + expf(-z));          // silu(z)
      act[(size_t)t * D_INNER + h * HEADDIM + d] = yv * sz;
    }
    px = x;
    WAIT_ASYNC();        // my async writes for step t+1 landed in LDS
    __syncthreads();     // everyone's writes visible; readers of S done
  }
}

// ======================================================================
// launch
// ======================================================================
extern "C" void kernel_launch(void* const* d_in, const int* in_sizes, int n_in,
                              void* d_out, int out_size, void* d_ws, size_t ws_size,
                              hipStream_t stream) {
  const float* u       = (const float*)d_in[0];   // [1024,1024]
  const float* w_in    = (const float*)d_in[1];   // [4480,1024]
  const float* w_out   = (const float*)d_in[2];   // [1024,2048]
  const float* dt_bias = (const float*)d_in[3];   // [32]
  const float* Dvec    = (const float*)d_in[4];   // [32]
  float* out = (float*)d_out;                     // [1024,1024]

  // workspace layout (floats)
  float* ws     = (float*)d_ws;
  float* proj   = ws;                                      // 1024*4480
  float* dtb    = proj   + (size_t)SEQLEN * DPROJ;         // 1024*32
  float* alphab = dtb    + (size_t)SEQLEN * NHEADS;
  float* betab  = alphab + (size_t)SEQLEN * NHEADS;
  float* gammab = betab  + (size_t)SEQLEN * NHEADS;
  float* cosb   = gammab + (size_t)SEQLEN * NHEADS;        // 1024*1024
  float* sinb   = cosb   + (size_t)SEQLEN * NHEADS * RROT;
  float* actb   = sinb   + (size_t)SEQLEN * NHEADS * RROT; // 1024*2048

  // 1) proj = u @ w_in^T : M=1024, N=4480, K=1024
  wmma_gemm_tn<<<dim3(SEQLEN / 16, DPROJ / 64), 128, 0, stream>>>(
      u, w_in, proj, D_MODEL, DPROJ);

  // 2) per-(t,h) gate scalars
  prep_scalars<<<(SEQLEN * NHEADS + 255) / 256, 256, 0, stream>>>(
      proj, dt_bias, dtb, alphab, betab, gammab);

  // 3) theta cumsum -> cos/sin tables
  theta_kernel<<<(NHEADS * RROT + 255) / 256, 256, 0, stream>>>(
      proj, dtb, cosb, sinb);

  // 4) sequential scan + fused D*x and SiLU gating -> act
  scan_kernel<<<NHEADS, 256, 0, stream>>>(
      proj, alphab, betab, gammab, cosb, sinb, Dvec, actb);

  // 5) out = act @ w_out^T : M=1024, N=1024, K=2048
  wmma_gemm_tn<<<dim3(SEQLEN / 16, D_MODEL / 64), 128, 0, stream>>>(
      actb, w_out, out, D_INNER, D_MODEL);
}